// MPS_30683246362894
// MI455X (gfx1250) — compile-verified
//
#include <hip/hip_runtime.h>

typedef __attribute__((ext_vector_type(16))) _Float16 v16h;
typedef __attribute__((ext_vector_type(8)))  _Float16 v8h;

#define N_SITES 64
#define N_MID   62
#define BOND    128
#define BATCH   16384

typedef union { uint4 q[2]; v16h h; } fragU;
typedef union { v8h  p[2]; v16h h; } pairU;

// ---------------------------------------------------------------------------
// Prologue: pack w_mid [62][128][2][128] f32 into f16 WMMA A-operand fragments
// of W^T in contiguous k order (elem s of lane half h holds d_in = 32f+16h+s),
// while the main kernel's C/D -> B-operand repack gives table k order.
//
// Fragment q = ((s*2 + i)*32 + mo*4 + f) occupies 1 KB, stored as TWO 512 B
// blocks:  halves j=0..7 at  q*1024 + lane*16
//          halves j=8..15 at q*1024 + 512 + lane*16
// so each ds_load_b128 in the main kernel has consecutive lanes reading
// consecutive 16 B chunks -> all 64 LDS banks hit exactly twice (no conflict).
// ---------------------------------------------------------------------------
__global__ void mps_pack_w(const float* __restrict__ w_mid,
                           _Float16* __restrict__ wpack) {
  const int gid  = blockIdx.x * blockDim.x + threadIdx.x;
  const int lane = gid & 31;
  const int frag = gid >> 5;
  const int f  = frag & 3;          // k-block of 32 (d_in)
  const int mo = (frag >> 2) & 7;   // m-block of 16 (d_out)
  const int i  = (frag >> 5) & 1;   // which of the two site matrices
  const int s  = frag >> 6;         // middle-site index
  if (s >= N_MID) return;
  const int h = lane >> 4;
  const int m = lane & 15;
  v8h lo, hi;
#pragma unroll
  for (int j = 0; j < 8; ++j) {
    const int d_out = 16 * mo + m;
    lo[j] = (_Float16)w_mid[((size_t)(s * BOND + (32 * f + 16 * h + j))     * 2 + i) * BOND + d_out];
    hi[j] = (_Float16)w_mid[((size_t)(s * BOND + (32 * f + 16 * h + 8 + j)) * 2 + i) * BOND + d_out];
  }
  v8h* out = (v8h*)wpack;                       // 16 B units
  out[(size_t)frag * 64 + lane]      = lo;
  out[(size_t)frag * 64 + 32 + lane] = hi;
}

// ---------------------------------------------------------------------------
// Main chain kernel: 128 threads = 4 waves; each wave owns TWO 16-row batch
// tiles (32 rows) and carries both states as f16 C/D tiles (8 x v8h each) for
// all 62 steps.  Per step: stage the step's 64 KB of fragments into LDS once
// per WG, then run 8 m-blocks x 4 k-blocks x {2 matrices x 2 batch tiles}
// = 128 v_wmma_f16_16x16x32_f16; every A fragment read from LDS feeds 2
// independent WMMA chains, then the per-batch scalars combine in packed f16.
// ---------------------------------------------------------------------------
__global__ void __launch_bounds__(128)
mps_chain(const float* __restrict__ x,
          const float* __restrict__ w_first,
          const _Float16* __restrict__ wpack,
          const float* __restrict__ w_last,
          float* __restrict__ out) {
  __shared__ uint4 smem[4096];                  // 64 KB: one step's fragments

  const int tid  = threadIdx.x;
  const int wave = tid >> 5;
  const int lane = tid & 31;
  const int n    = lane & 15;                   // batch column within tile
  const int b0   = blockIdx.x * 128 + wave * 32 + n;
  const int b1   = b0 + 16;

  // ---- site 0 init.  f16 state layout: tile t, half h, elem j <-> d = 16t+8h+j.
  v8h accA[8], accB[8];
#pragma unroll
  for (int t = 0; t < 8; ++t) {
#pragma unroll
    for (int j = 0; j < 8; ++j) { accA[t][j] = (_Float16)0.0f; accB[t][j] = (_Float16)0.0f; }
  }
  {
    const float2 xa = *(const float2*)(x + (size_t)b0 * 128);
    const float2 xb = *(const float2*)(x + (size_t)b1 * 128);
    if (lane < 16) {                            // d = 0,1 live in half h = 0
      accA[0][0] = (_Float16)(w_first[0] * xa.x + w_first[2] * xa.y);
      accA[0][1] = (_Float16)(w_first[1] * xa.x + w_first[3] * xa.y);
      accB[0][0] = (_Float16)(w_first[0] * xb.x + w_first[2] * xb.y);
      accB[0][1] = (_Float16)(w_first[1] * xb.x + w_first[3] * xb.y);
    }
  }

  const uint4* wp4 = (const uint4*)wpack;

  for (int s = 0; s < N_MID; ++s) {
    // ---- stage this step's 64 KB into LDS (512 B per thread, coalesced,
    // conflict-free: consecutive lanes touch consecutive 16 B chunks).
    {
      const uint4* src = wp4 + (size_t)s * 4096;
#pragma unroll
      for (int u = 0; u < 32; ++u)
        smem[u * 128 + tid] = src[u * 128 + tid];
      if (s + 1 < N_MID) {                      // prefetch next step's weights
        const char* nxt = (const char*)(wp4 + (size_t)(s + 1) * 4096);
#pragma unroll
        for (int v = 0; v < 4; ++v)
          __builtin_prefetch(nxt + tid * 512 + v * 128, 0, 3);
      }
    }
    __syncthreads();

    // per-batch scalars for this site (lane-private: batch == N column)
    const float2 xA = *(const float2*)(x + (size_t)b0 * 128 + (size_t)(s + 1) * 2);
    const float2 xB = *(const float2*)(x + (size_t)b1 * 128 + (size_t)(s + 1) * 2);
    const _Float16 xA0 = (_Float16)xA.x, xA1 = (_Float16)xA.y;
    const _Float16 xB0 = (_Float16)xB.x, xB1 = (_Float16)xB.y;

    // ---- C/D -> B operand: pure register concat (table k order, no cvt)
    pairU bfA[4], bfB[4];
#pragma unroll
    for (int f = 0; f < 4; ++f) {
      bfA[f].p[0] = accA[2 * f];  bfA[f].p[1] = accA[2 * f + 1];
      bfB[f].p[0] = accB[2 * f];  bfB[f].p[1] = accB[2 * f + 1];
    }

    // ---- 8 x 4 x (2 matrices x 2 batch tiles) WMMAs, pipelined LDS loads
    fragU a0[2], a1[2];
    a0[0].q[0] = smem[lane];            a0[0].q[1] = smem[32 + lane];
    a1[0].q[0] = smem[32 * 64 + lane];  a1[0].q[1] = smem[32 * 64 + 32 + lane];

#pragma unroll
    for (int mo = 0; mo < 8; ++mo) {
      v8h e00, e01, e10, e11;
#pragma unroll
      for (int j = 0; j < 8; ++j) {
        e00[j] = (_Float16)0.0f; e01[j] = (_Float16)0.0f;
        e10[j] = (_Float16)0.0f; e11[j] = (_Float16)0.0f;
      }
#pragma unroll
      for (int f = 0; f < 4; ++f) {
        const int q   = mo * 4 + f;
        const int cur = q & 1;
        const int nq  = q + 1;
        if (nq < 32) {                          // prefetch next fragment pair
          const int nb0 = nq * 64;
          const int nb1 = (32 + nq) * 64;
          a0[nq & 1].q[0] = smem[nb0 + lane];
          a0[nq & 1].q[1] = smem[nb0 + 32 + lane];
          a1[nq & 1].q[0] = smem[nb1 + lane];
          a1[nq & 1].q[1] = smem[nb1 + 32 + lane];
        }
        // 4 independent chains; each A fragment feeds 2 (2x LDS reuse)
        e00 = __builtin_amdgcn_wmma_f16_16x16x32_f16(
                  false, a0[cur].h, false, bfA[f].h, (short)0, e00, false, false);
        e01 = __builtin_amdgcn_wmma_f16_16x16x32_f16(
                  false, a0[cur].h, false, bfB[f].h, (short)0, e01, false, false);
        e10 = __builtin_amdgcn_wmma_f16_16x16x32_f16(
                  false, a1[cur].h, false, bfA[f].h, (short)0, e10, false, false);
        e11 = __builtin_amdgcn_wmma_f16_16x16x32_f16(
                  false, a1[cur].h, false, bfB[f].h, (short)0, e11, false, false);
      }
#pragma unroll
      for (int j = 0; j < 8; ++j) {             // packed f16 combine
        accA[mo][j] = (_Float16)(xA0 * e00[j] + xA1 * e10[j]);
        accB[mo][j] = (_Float16)(xB0 * e01[j] + xB1 * e11[j]);
      }
    }
    __syncthreads();                            // LDS reuse next iteration
  }

  // ---- last site
  if (lane < 16) {
    const float2 xa = *(const float2*)(x + (size_t)b0 * 128 + 63 * 2);
    const float2 xb = *(const float2*)(x + (size_t)b1 * 128 + 63 * 2);
    const float cA0 = w_last[0] * xa.x + w_last[1] * xa.y;
    const float cA1 = w_last[2] * xa.x + w_last[3] * xa.y;
    const float cB0 = w_last[0] * xb.x + w_last[1] * xb.y;
    const float cB1 = w_last[2] * xb.x + w_last[3] * xb.y;
    out[b0] = (float)accA[0][0] * cA0 + (float)accA[0][1] * cA1;
    out[b1] = (float)accB[0][0] * cB0 + (float)accB[0][1] * cB1;
  }
}

extern "C" void kernel_launch(void* const* d_in, const int* in_sizes, int n_in,
                              void* d_out, int out_size, void* d_ws, size_t ws_size,
                              hipStream_t stream) {
  const float* x       = (const float*)d_in[0];
  const float* w_first = (const float*)d_in[1];
  const float* w_mid   = (const float*)d_in[2];
  const float* w_last  = (const float*)d_in[3];
  float* out           = (float*)d_out;
  _Float16* wpack      = (_Float16*)d_ws;       // needs 62*64KB ~= 3.9 MB

  // 62 steps * 2 matrices * 32 fragments * 32 lanes threads = 126976
  const int pack_threads = N_MID * 2 * 32 * 32;
  mps_pack_w<<<pack_threads / 256, 256, 0, stream>>>(w_mid, wpack);

  // 16384 batch / (4 waves * 32 rows) = 128 blocks of 128 threads
  mps_chain<<<BATCH / 128, 128, 0, stream>>>(x, w_first, wpack, w_last, out);
}